// PNG_63247688401062
// MI455X (gfx1250) — compile-verified
//
#include <hip/hip_runtime.h>
#include <hip/hip_bf16.h>

typedef __attribute__((ext_vector_type(2))) float floatx2;
typedef __attribute__((ext_vector_type(8))) float floatx8;

// address-space pointer types for the gfx1250 async global->LDS builtin
typedef __attribute__((address_space(1))) int as1_int;
typedef __attribute__((address_space(3))) int as3_int;

#define U_DIM   8192
#define I_DIM   8192
#define R_TOTAL 16384   // user rows then item rows, one uniform problem

// Detect gfx1250 async global->LDS builtins (guarded: build never breaks)
#if defined(__has_builtin)
#if __has_builtin(__builtin_amdgcn_global_load_async_to_lds_b32)
#define HAVE_ASYNC_LDS 1
#endif
#if __has_builtin(__builtin_amdgcn_s_wait_asynccnt)
#define HAVE_WAIT_ASYNC_BUILTIN 1
#endif
#endif

__device__ __forceinline__ void wait_asynccnt0() {
#if defined(HAVE_WAIT_ASYNC_BUILTIN)
  __builtin_amdgcn_s_wait_asynccnt(0);
#else
  asm volatile("s_wait_asynccnt 0" ::: "memory");
#endif
}

// reversed db3 analysis taps (cross-correlation kernel == true convolution),
// produced as a branch-free select chain over immediates (no memory, no branches)
__device__ __forceinline__ float dbtap(int d, bool hi) {
  float r = 0.f;
  if (!hi) {
    r = (d == 0) ? 0.3326705529509569f   : r;
    r = (d == 1) ? 0.8068915093133388f   : r;
    r = (d == 2) ? 0.4598775021193313f   : r;
    r = (d == 3) ? -0.13501102001039084f : r;
    r = (d == 4) ? -0.08544127388224149f : r;
    r = (d == 5) ? 0.035226291882100656f : r;
  } else {
    r = (d == 0) ? 0.035226291882100656f : r;
    r = (d == 1) ? 0.08544127388224149f  : r;
    r = (d == 2) ? -0.13501102001039084f : r;
    r = (d == 3) ? -0.4598775021193313f  : r;
    r = (d == 4) ? 0.8068915093133388f   : r;
    r = (d == 5) ? -0.3326705529509569f  : r;
  }
  return r;
}

// ---------------------------------------------------------------------------
// Scatter / normalization kernels (bandwidth-trivial vs the dense stage)
// ---------------------------------------------------------------------------
__global__ void k_count(const int* __restrict__ item, int nnz, float* __restrict__ counts) {
  int i = blockIdx.x * 256 + threadIdx.x;
  if (i < nnz) atomicAdd(&counts[item[i]], 1.0f);
}

__global__ void k_popscat(const int* __restrict__ item, const float* __restrict__ val,
                          int nnz, const float* __restrict__ counts, float* __restrict__ pop) {
  int i = blockIdx.x * 256 + threadIdx.x;
  if (i < nnz) {
    int it = item[i];
    atomicAdd(&pop[it], val[i] * log1pf(counts[it]));
  }
}

__global__ void k_sumsq(const float* __restrict__ x, int n, float* __restrict__ acc) {
  __shared__ float red[256];
  float s = 0.f;
  for (int i = blockIdx.x * 256 + threadIdx.x; i < n; i += gridDim.x * 256) s += x[i] * x[i];
  red[threadIdx.x] = s;
  __syncthreads();
  for (int o = 128; o > 0; o >>= 1) {
    if (threadIdx.x < o) red[threadIdx.x] += red[threadIdx.x + o];
    __syncthreads();
  }
  if (threadIdx.x == 0) atomicAdd(acc, red[0]);
}

__global__ void k_scale(float* __restrict__ x, int n, const float* __restrict__ acc) {
  int i = blockIdx.x * 256 + threadIdx.x;
  if (i < n) x[i] *= 1.0f / (sqrtf(*acc) + 1e-8f);
}

__global__ void k_actscat(const int* __restrict__ user, const int* __restrict__ item,
                          const float* __restrict__ val, int nnz,
                          const float* __restrict__ pop, float* __restrict__ act) {
  int i = blockIdx.x * 256 + threadIdx.x;
  if (i < nnz) {
    atomicAdd(&act[user[i]], val[i] / log1pf(pop[item[i]] + 1e-8f));
  }
}

__global__ void k_sigscat(const int* __restrict__ user, const int* __restrict__ item,
                          const float* __restrict__ val, int nnz, float* __restrict__ sig) {
  int i = blockIdx.x * 256 + threadIdx.x;
  if (i < nnz) {
    int u = user[i], it = item[i];
    float v = val[i];
    atomicAdd(&sig[(long long)u * I_DIM + it], v);                    // user_sig row
    atomicAdd(&sig[(long long)(U_DIM + it) * U_DIM + u], v);          // item_sig row
  }
}

// ---------------------------------------------------------------------------
// One DWT analysis level via WMMA F32 16x16x4.
// Tile: 16 rows x 16 outputs. D(16x16) = A(16x36) x B(36x16), B banded with
// B[i][j] = K[i-2j] (stride-2 conv, 6 taps, left pad 4). 9 WMMA K-steps.
// Both banded B matrices live in LDS (built once per block, branch-free), so
// the steady state per K-step is: 3x ds_load_2addr + 2x v_wmma, no branches.
// ---------------------------------------------------------------------------
__global__ __launch_bounds__(256) void k_dwt(
    const float* __restrict__ in, int instride, int N,
    float* __restrict__ out, int ostride, int outsize,
    const float* __restrict__ act, const float* __restrict__ pop, int useScale,
    float* __restrict__ hisum) {
  __shared__ float sm[16 * 273];     // 16 rows x 260 window, stride 273 (conflict-free)
  __shared__ float sBlo[36 * 16];    // banded filter matrices B[i][jj]
  __shared__ float sBhi[36 * 16];

  const int tid = threadIdx.x;
  const int r0  = blockIdx.y * 16;
  const int J0  = blockIdx.x * 128;

  // build banded B matrices once (pure VALU selects over immediates)
  for (int idx = tid; idx < 36 * 16; idx += 256) {
    int i = idx >> 4, jj = idx & 15;
    int d = i - 2 * jj;
    bool ok = (d >= 0) && (d < 6);
    sBlo[idx] = ok ? dbtap(d, false) : 0.f;
    sBhi[idx] = ok ? dbtap(d, true)  : 0.f;
  }

  // fill the shared input window (coalesced); async global->LDS when no
  // row-scaling is fused (levels 2,3) and the builtin is available
  for (int idx = tid; idx < 16 * 260; idx += 256) {
    int r = idx / 260, i = idx - r * 260;
    int gc = 2 * J0 - 4 + i;
    int gr = r0 + r;
    float* dst = &sm[r * 273 + i];
#if defined(HAVE_ASYNC_LDS)
    if (!useScale) {
      if (gc >= 0 && gc < N) {
        __builtin_amdgcn_global_load_async_to_lds_b32(
            (as1_int*)(in + (long long)gr * instride + gc),
            (as3_int*)dst, 0, 0);
      } else {
        *dst = 0.f;
      }
      continue;
    }
#endif
    float v = 0.f;
    if (gc >= 0 && gc < N) v = in[(long long)gr * instride + gc];
    if (useScale) v *= (gr < U_DIM) ? (1.f + act[gr]) : (1.f + pop[gr - U_DIM]);
    *dst = v;
  }
#if defined(HAVE_ASYNC_LDS)
  if (!useScale) wait_asynccnt0();
#endif
  __syncthreads();

  const int wv   = tid >> 5;        // wave id: 8 output tiles per block
  const int lane = tid & 31;
  const int j0   = J0 + wv * 16;
  if (j0 >= outsize) return;        // wave-uniform exit, EXEC stays all-ones

  const int jj   = lane & 15;
  const int hi16 = lane >> 4;       // 0: K-phases {0,1}; 1: K-phases {2,3}
  const float* aRow = &sm[jj * 273 + 32 * wv];   // A row base for this lane
  const int   bOff  = 2 * hi16 * 16 + jj;        // B base for this lane

  floatx8 clo = {};
  floatx8 chi = {};
#pragma unroll
  for (int s = 0; s < 9; ++s) {
    const int ka = 4 * s + 2 * hi16;
    floatx2 a;                               // A: row jj, K cols ka / ka+1
    a.x = aRow[ka];
    a.y = aRow[ka + 1];
    const int bb = 4 * s * 16 + bOff;        // B: rows ka/ka+1, col jj
    floatx2 blo, bhi;
    blo.x = sBlo[bb];
    blo.y = sBlo[bb + 16];
    bhi.x = sBhi[bb];
    bhi.y = sBhi[bb + 16];
    clo = __builtin_amdgcn_wmma_f32_16x16x4_f32(false, a, false, blo, (short)0, clo, false, false);
    chi = __builtin_amdgcn_wmma_f32_16x16x4_f32(false, a, false, bhi, (short)0, chi, false, false);
  }

  const bool colok = (j0 + jj) < outsize;
#pragma unroll
  for (int v = 0; v < 8; ++v) {
    int gr = r0 + v + hi16 * 8;             // C/D layout: VGPR v -> row v (+8 for lanes>=16)
    if (colok) out[(long long)gr * ostride + j0 + jj] = clo[v];
    // per-row sum of hi^2 across the 16 lanes holding that row
    float h = colok ? chi[v] * chi[v] : 0.f;
    h += __shfl_xor(h, 1);
    h += __shfl_xor(h, 2);
    h += __shfl_xor(h, 4);
    h += __shfl_xor(h, 8);
    if (jj == 0) atomicAdd(&hisum[gr], h);
  }
}

// mean of the final low-pass row (1028 valid cols, stride 1040)
__global__ void k_mean(const float* __restrict__ lo3, float* __restrict__ low) {
  __shared__ float red[256];
  int r = blockIdx.x;
  float s = 0.f;
  for (int c = threadIdx.x; c < 1028; c += 256) s += lo3[(long long)r * 1040 + c];
  red[threadIdx.x] = s;
  __syncthreads();
  for (int o = 128; o > 0; o >>= 1) {
    if (threadIdx.x < o) red[threadIdx.x] += red[threadIdx.x + o];
    __syncthreads();
  }
  if (threadIdx.x == 0) low[r] = red[0] / 1028.0f;
}

// Dirichlet-expectation weight per row
__global__ void k_rows(const float* __restrict__ low, const float* __restrict__ hisum,
                       float* __restrict__ wpop) {
  int r = blockIdx.x * 256 + threadIdx.x;
  if (r < R_TOTAL) {
    float hs = sqrtf(hisum[r]) + sqrtf(hisum[R_TOTAL + r]) + sqrtf(hisum[2 * R_TOTAL + r]);
    float a1 = fmaxf(low[r], 1e-6f);
    float a2 = fmaxf(hs, 1e-6f);
    wpop[r] = a1 / (a1 + a2);
  }
}

// final nnz-valued outputs: [4, NNZ]
__global__ void k_out(const int* __restrict__ user, const int* __restrict__ item,
                      const float* __restrict__ val, int nnz,
                      const float* __restrict__ wpop, float* __restrict__ out) {
  int i = blockIdx.x * 256 + threadIdx.x;
  if (i < nnz) {
    float v = val[i];
    float wu = wpop[user[i]];
    float wi = wpop[U_DIM + item[i]];
    float wun = 1.f - wu, win = 1.f - wi;
    out[i]            = v * wu  * wi;
    out[nnz + i]      = v * wun * win;
    out[2LL*nnz + i]  = v * wu  * win;
    out[3LL*nnz + i]  = v * wun * wi;
  }
}

// ---------------------------------------------------------------------------
extern "C" void kernel_launch(void* const* d_in, const int* in_sizes, int n_in,
                              void* d_out, int out_size, void* d_ws, size_t ws_size,
                              hipStream_t stream) {
  const int*   user = (const int*)d_in[0];
  const int*   item = (const int*)d_in[1];
  const float* val  = (const float*)d_in[2];
  const int    nnz  = in_sizes[0];
  float*       out  = (float*)d_out;

  char* ws = (char*)d_ws;
  // small arrays
  float* counts = (float*)(ws + 0);            //  8192 f
  float* pop    = (float*)(ws + 32768);        //  8192 f
  float* act    = (float*)(ws + 65536);        //  8192 f
  float* scal   = (float*)(ws + 98304);        //  2 f   (pop sumsq, act sumsq)
  float* hisum  = (float*)(ws + 98560);        //  3*16384 f
  float* low    = (float*)(ws + 295168);       //  16384 f
  float* wpop   = (float*)(ws + 360704);       //  16384 f
  // big buffers
  float* sig = (float*)(ws + 524288);                         // 16384 x 8192   (512 MB)
  float* lo1 = (float*)(ws + 524288 + 536870912ULL);          // 16384 x 4112   (~270 MB)
  float* lo2 = (float*)(ws + 524288);                         // reuse sig region (dead after L1)
  float* lo3 = (float*)(ws + 524288 + 135266304ULL);          // after lo2, inside sig region

  const int gn = (nnz + 255) / 256;

  // zero: small arrays + hisum, and the dense scatter target
  (void)hipMemsetAsync(ws, 0, 295168, stream);
  (void)hipMemsetAsync(sig, 0, 536870912ULL, stream);

  // popularity / activity
  k_count  <<<gn, 256, 0, stream>>>(item, nnz, counts);
  k_popscat<<<gn, 256, 0, stream>>>(item, val, nnz, counts, pop);
  k_sumsq  <<<64, 256, 0, stream>>>(pop, U_DIM, &scal[0]);
  k_scale  <<<32, 256, 0, stream>>>(pop, I_DIM, &scal[0]);
  k_actscat<<<gn, 256, 0, stream>>>(user, item, val, nnz, pop, act);
  k_sumsq  <<<64, 256, 0, stream>>>(act, U_DIM, &scal[1]);
  k_scale  <<<32, 256, 0, stream>>>(act, U_DIM, &scal[1]);

  // dense signal matrices (user rows 0..8191, item rows 8192..16383)
  k_sigscat<<<gn, 256, 0, stream>>>(user, item, val, nnz, sig);

  // 3 DWT levels via WMMA f32 16x16x4  (row scaling fused into level 1)
  k_dwt<<<dim3(33, R_TOTAL / 16), 256, 0, stream>>>(sig, 8192, 8192, lo1, 4112, 4098,
                                                    act, pop, 1, hisum);
  k_dwt<<<dim3(17, R_TOTAL / 16), 256, 0, stream>>>(lo1, 4112, 4098, lo2, 2064, 2051,
                                                    act, pop, 0, hisum + R_TOTAL);
  k_dwt<<<dim3(9,  R_TOTAL / 16), 256, 0, stream>>>(lo2, 2064, 2051, lo3, 1040, 1028,
                                                    act, pop, 0, hisum + 2 * R_TOTAL);

  // per-row features -> weights -> outputs
  k_mean<<<R_TOTAL, 256, 0, stream>>>(lo3, low);
  k_rows<<<R_TOTAL / 256, 256, 0, stream>>>(low, hisum, wpop);
  k_out <<<gn, 256, 0, stream>>>(user, item, val, nnz, wpop, out);
}